// MoRKVCache_17317308138095
// MI455X (gfx1250) — compile-verified
//
#include <hip/hip_runtime.h>

typedef __attribute__((ext_vector_type(2))) float v2f;
typedef __attribute__((ext_vector_type(8))) float v8f;

#define Bsz  4
#define Sseq 4096
#define Hd   1024
#define Tt   8
#define Cc   256
#define ROWS (Bsz * Sseq)      // 16384 output rows
#define PADH 1040              // padded LDS row: lane halves hit disjoint banks

// out[row, h] = sum_{t<8} rw[row, t] * cache[t, (pos[t]+1)%C, h]
// Done as (16x8) x (8x16) tiles via two chained V_WMMA_F32_16X16X4_F32.
__global__ __launch_bounds__(256) void mor_retrieve_kernel(
    const float* __restrict__ rw,     // [ROWS, T]
    const float* __restrict__ kc,     // [T, C, H]
    const float* __restrict__ vc,     // [T, C, H]
    const int*   __restrict__ cpos,   // [T]
    float* __restrict__ outK,         // [ROWS, H]
    float* __restrict__ outV)         // [ROWS, H]
{
    __shared__ float s_k[Tt * PADH];
    __shared__ float s_v[Tt * PADH];

    const int tid = threadIdx.x;

    int rp[Tt];
#pragma unroll
    for (int t = 0; t < Tt; ++t) rp[t] = (cpos[t] + 1) & (Cc - 1);

    // Stage recent_keys / recent_values [T][H] into LDS (float4 per load).
    for (int i = tid; i < Tt * Hd / 4; i += 256) {
        const int t  = i >> 8;            // 256 float4s per H-row
        const int h4 = (i & 255) << 2;
        const size_t g = ((size_t)t * Cc + rp[t]) * Hd + h4;
        const float4 kk = *(const float4*)(kc + g);
        const float4 vv = *(const float4*)(vc + g);
        *(float4*)(s_k + t * PADH + h4) = kk;
        *(float4*)(s_v + t * PADH + h4) = vv;
    }
    __syncthreads();

    const int wave = tid >> 5;
    const int lane = tid & 31;
    const int m    = lane & 15;       // row (A) / col (B,D) within tile
    const int half = lane >> 4;       // 0: K 0,1 | 1: K 2,3 (A,B frags)
    const int kb   = half << 1;

#pragma unroll
    for (int rt = 0; rt < 4; ++rt) {
        const int row0 = (blockIdx.x * 4 + rt) << 4;

        // A fragment: 16x4 f32 tile of rw, K = t. lane<16 -> K 0,1 ; lane>=16 -> K 2,3.
        const float* arow = rw + (size_t)(row0 + m) * Tt;
        v2f a0, a1;
        a0.x = arow[kb];     a0.y = arow[kb + 1];      // t = kb, kb+1
        a1.x = arow[kb + 4]; a1.y = arow[kb + 5];      // t = kb+4, kb+5

        for (int i = 0; i < 8; ++i) {
            const int h0  = ((wave << 3) + i) << 4;
            const int col = h0 + m;

            // B fragments: 4x16 rows of recent[t][col]. VGPR0 = K {0|2}, VGPR1 = K {1|3}.
            v2f bk0, bk1, bv0, bv1;
            bk0.x = s_k[ kb      * PADH + col];
            bk0.y = s_k[(kb + 1) * PADH + col];
            bk1.x = s_k[(kb + 4) * PADH + col];
            bk1.y = s_k[(kb + 5) * PADH + col];
            bv0.x = s_v[ kb      * PADH + col];
            bv0.y = s_v[(kb + 1) * PADH + col];
            bv1.x = s_v[(kb + 4) * PADH + col];
            bv1.y = s_v[(kb + 5) * PADH + col];

            v8f ck = {};
            ck = __builtin_amdgcn_wmma_f32_16x16x4_f32(false, a0, false, bk0, (short)0, ck, false, false);
            ck = __builtin_amdgcn_wmma_f32_16x16x4_f32(false, a1, false, bk1, (short)0, ck, false, false);
            v8f cv = {};
            cv = __builtin_amdgcn_wmma_f32_16x16x4_f32(false, a0, false, bv0, (short)0, cv, false, false);
            cv = __builtin_amdgcn_wmma_f32_16x16x4_f32(false, a1, false, bv1, (short)0, cv, false, false);

            // D: VGPR r -> row r (lanes 0-15) / r+8 (lanes 16-31), col = h0 + m.
            float* pk = outK + (size_t)(row0 + half * 8) * Hd + col;
            float* pv = outV + (size_t)(row0 + half * 8) * Hd + col;
#pragma unroll
            for (int r = 0; r < 8; ++r) {
                __builtin_nontemporal_store(ck[r], pk + (size_t)r * Hd);
                __builtin_nontemporal_store(cv[r], pv + (size_t)r * Hd);
            }
        }
    }
}

extern "C" void kernel_launch(void* const* d_in, const int* in_sizes, int n_in,
                              void* d_out, int out_size, void* d_ws, size_t ws_size,
                              hipStream_t stream) {
    // setup_inputs order: keys(0), values(1), routing_weights(2),
    //                     key_cache(3), value_cache(4), cache_positions(5).
    // keys/values are dead for the returned tuple (write_pos != read_pos always).
    const float* rw  = (const float*)d_in[2];
    const float* kc  = (const float*)d_in[3];
    const float* vc  = (const float*)d_in[4];
    const int*   pos = (const int*)d_in[5];

    float* outK = (float*)d_out;                       // retrieved_keys  [B,S,H]
    float* outV = outK + (size_t)ROWS * Hd;            // retrieved_values [B,S,H]

    mor_retrieve_kernel<<<dim3(ROWS / 16 / 4), dim3(256), 0, stream>>>(
        rw, kc, vc, pos, outK, outV);
}